// GCN_32409823216071
// MI455X (gfx1250) — compile-verified
//
#include <hip/hip_runtime.h>
#include <math.h>

#define N_NODES 10000
#define N_EDGES 320000
#define NFEAT   512
#define NHID    256
#define NCLASS  40

typedef float v2f __attribute__((ext_vector_type(2)));
typedef float v8f __attribute__((ext_vector_type(8)));

// ---------------------------------------------------------------- utilities
__global__ void gcn_zero_kernel(float* __restrict__ p, int n) {
    int i = blockIdx.x * blockDim.x + threadIdx.x;
    if (i < n) p[i] = 0.0f;
}

// ---------------------------------------------------------------- GEMM1: XW = X[10000,512] @ W1[512,256]
// Block = 256 threads = 8 waves. Block handles one 16-row M tile and all N=256.
// Wave w computes 16x32 (two 16x16 WMMA accumulators) at columns [w*32, w*32+32).
__global__ __launch_bounds__(256) void gcn_gemm1_kernel(
    const float* __restrict__ X, const float* __restrict__ W,
    float* __restrict__ XW) {
    const int wave = threadIdx.x >> 5;
    const int lane = threadIdx.x & 31;
    const int m0   = blockIdx.x * 16;

    const int lrow  = lane & 15;          // row within tile (A and D col index)
    const int half  = lane >> 4;          // 0: K 0..1 / rows 0..7, 1: K 2..3 / rows 8..15
    const int khalf = half * 2;

    const int col0 = wave * 32 + lrow;
    const int col1 = col0 + 16;

    const float* __restrict__ arow = X + (size_t)(m0 + lrow) * NFEAT;

    v8f acc0 = {};
    v8f acc1 = {};

#pragma unroll 4
    for (int k = 0; k < NFEAT; k += 4) {
        v2f a;
        a.x = arow[k + khalf];
        a.y = arow[k + khalf + 1];

        const float* __restrict__ bk = W + (size_t)(k + khalf) * NHID;
        v2f b0, b1;
        b0.x = bk[col0];
        b0.y = bk[NHID + col0];
        b1.x = bk[col1];
        b1.y = bk[NHID + col1];

        acc0 = __builtin_amdgcn_wmma_f32_16x16x4_f32(false, a, false, b0,
                                                     (short)0, acc0, false, false);
        acc1 = __builtin_amdgcn_wmma_f32_16x16x4_f32(false, a, false, b1,
                                                     (short)0, acc1, false, false);
    }

    // D layout: VGPR r -> row m0 + r + 8*half, col = lane%16
    const int mbase = m0 + 8 * half;
#pragma unroll
    for (int r = 0; r < 8; ++r) {
        const size_t m = (size_t)(mbase + r);
        XW[m * NHID + col0] = acc0[r];
        XW[m * NHID + col1] = acc1[r];
    }
}

// ---------------------------------------------------------------- GEMM2: HW = H[10000,256] @ W2[256,40]
// Block = 96 threads = 3 waves; wave w covers N tile [w*16, w*16+16), cols >= 40 masked.
__global__ __launch_bounds__(96) void gcn_gemm2_kernel(
    const float* __restrict__ H, const float* __restrict__ W2,
    float* __restrict__ HW) {
    const int wave = threadIdx.x >> 5;    // 0..2
    const int lane = threadIdx.x & 31;
    const int m0   = blockIdx.x * 16;

    const int lrow  = lane & 15;
    const int half  = lane >> 4;
    const int khalf = half * 2;

    const int col   = wave * 16 + lrow;
    const bool cok  = (col < NCLASS);

    const float* __restrict__ arow = H + (size_t)(m0 + lrow) * NHID;

    v8f acc = {};
#pragma unroll 4
    for (int k = 0; k < NHID; k += 4) {
        v2f a;
        a.x = arow[k + khalf];
        a.y = arow[k + khalf + 1];
        v2f b;
        b.x = cok ? W2[(size_t)(k + khalf) * NCLASS + col] : 0.0f;
        b.y = cok ? W2[(size_t)(k + khalf + 1) * NCLASS + col] : 0.0f;
        acc = __builtin_amdgcn_wmma_f32_16x16x4_f32(false, a, false, b,
                                                    (short)0, acc, false, false);
    }

    if (cok) {
        const int mbase = m0 + 8 * half;
#pragma unroll
        for (int r = 0; r < 8; ++r) {
            HW[(size_t)(mbase + r) * NCLASS + col] = acc[r];
        }
    }
}

// ---------------------------------------------------------------- SpMM: out[dst] += w * Hin[src]
// One thread per (edge, 4 consecutive features); feature index fastest -> coalesced.
__global__ void gcn_spmm_kernel(const int* __restrict__ src,
                                const int* __restrict__ dst,
                                const float* __restrict__ w,
                                const float* __restrict__ Hin,
                                float* __restrict__ Hout,
                                int chunks, int D, int total) {
    int idx = blockIdx.x * blockDim.x + threadIdx.x;
    if (idx >= total) return;
    const int e  = idx / chunks;
    const int f  = (idx - e * chunks) * 4;
    const int s  = src[e];
    const int d  = dst[e];
    const float wt = w[e];
    const float4 v = *(const float4*)(Hin + (size_t)s * D + f);
    float* op = Hout + (size_t)d * D + f;
    atomicAdd(op + 0, wt * v.x);
    atomicAdd(op + 1, wt * v.y);
    atomicAdd(op + 2, wt * v.z);
    atomicAdd(op + 3, wt * v.w);
}

// ---------------------------------------------------------------- bias + ReLU, in place over [N_NODES, NHID]
__global__ void gcn_bias_relu_kernel(float* __restrict__ h,
                                     const float* __restrict__ b, int n) {
    int i = blockIdx.x * blockDim.x + threadIdx.x;
    if (i < n) {
        float v = h[i] + b[i & (NHID - 1)];
        h[i] = v > 0.0f ? v : 0.0f;
    }
}

// ---------------------------------------------------------------- bias + log_softmax, one wave32 per row of 40
__global__ __launch_bounds__(256) void gcn_logsoftmax_kernel(
    const float* __restrict__ acc, const float* __restrict__ b2,
    float* __restrict__ out) {
    const int row  = blockIdx.x * (blockDim.x >> 5) + (threadIdx.x >> 5);
    const int lane = threadIdx.x & 31;
    if (row >= N_NODES) return;

    const float* a = acc + (size_t)row * NCLASS;
    float x0 = a[lane] + b2[lane];                       // lane 0..31 < 40 always
    const bool hi = (lane + 32) < NCLASS;                // lanes 0..7 hold a 2nd element
    float x1 = hi ? (a[lane + 32] + b2[lane + 32]) : -INFINITY;

    float m = fmaxf(x0, x1);
#pragma unroll
    for (int off = 16; off > 0; off >>= 1)
        m = fmaxf(m, __shfl_xor(m, off, 32));

    float s = __expf(x0 - m) + (hi ? __expf(x1 - m) : 0.0f);
#pragma unroll
    for (int off = 16; off > 0; off >>= 1)
        s += __shfl_xor(s, off, 32);

    const float lse = m + __logf(s);
    out[(size_t)row * NCLASS + lane] = x0 - lse;
    if (hi) out[(size_t)row * NCLASS + lane + 32] = x1 - lse;
}

// ---------------------------------------------------------------- launch
extern "C" void kernel_launch(void* const* d_in, const int* in_sizes, int n_in,
                              void* d_out, int out_size, void* d_ws, size_t ws_size,
                              hipStream_t stream) {
    const float* x   = (const float*)d_in[0];
    const int*   esrc = (const int*)d_in[1];
    const int*   edst = (const int*)d_in[2];
    const float* ew  = (const float*)d_in[3];
    const float* W1  = (const float*)d_in[4];
    const float* b1  = (const float*)d_in[5];
    const float* W2  = (const float*)d_in[6];
    const float* b2  = (const float*)d_in[7];
    float* out = (float*)d_out;

    float* ws = (float*)d_ws;
    // workspace layout (floats)
    const size_t OFF_XW   = 0;                       // [10000, 256]
    const size_t OFF_ACC1 = OFF_XW   + (size_t)N_NODES * NHID;   // [10000, 256]
    const size_t OFF_ACC2 = OFF_ACC1 + (size_t)N_NODES * NHID;   // [10000, 40]
    const size_t OFF_HW   = OFF_ACC2 + (size_t)N_NODES * NCLASS; // [10000, 40]

    float* XW   = ws + OFF_XW;
    float* ACC1 = ws + OFF_ACC1;
    float* ACC2 = ws + OFF_ACC2;
    float* HW   = ws + OFF_HW;

    // 1) zero the two scatter accumulators (ACC1 and ACC2 are contiguous)
    {
        const int n = (int)(N_NODES * NHID + N_NODES * NCLASS);
        gcn_zero_kernel<<<(n + 255) / 256, 256, 0, stream>>>(ACC1, n);
    }

    // 2) XW = x @ W1   (WMMA f32 16x16x4)
    gcn_gemm1_kernel<<<N_NODES / 16, 256, 0, stream>>>(x, W1, XW);

    // 3) ACC1[dst] += w * XW[src]
    {
        const int chunks = NHID / 4;
        const int total  = N_EDGES * chunks;
        gcn_spmm_kernel<<<(total + 255) / 256, 256, 0, stream>>>(
            esrc, edst, ew, XW, ACC1, chunks, NHID, total);
    }

    // 4) H = relu(ACC1 + b1)  (in place)
    {
        const int n = N_NODES * NHID;
        gcn_bias_relu_kernel<<<(n + 255) / 256, 256, 0, stream>>>(ACC1, b1, n);
    }

    // 5) HW = H @ W2   (WMMA f32 16x16x4, N padded 40->48 with masking)
    gcn_gemm2_kernel<<<N_NODES / 16, 96, 0, stream>>>(ACC1, W2, HW);

    // 6) ACC2[dst] += w * HW[src]
    {
        const int chunks = NCLASS / 4;
        const int total  = N_EDGES * chunks;
        gcn_spmm_kernel<<<(total + 255) / 256, 256, 0, stream>>>(
            esrc, edst, ew, HW, ACC2, chunks, NCLASS, total);
    }

    // 7) out = log_softmax(ACC2 + b2) rowwise, one wave per row
    gcn_logsoftmax_kernel<<<(N_NODES + 7) / 8, 256, 0, stream>>>(ACC2, b2, out);
}